// RelativePositionalMultiHeadSelfAttention_22857815949928
// MI455X (gfx1250) — compile-verified
//
#include <hip/hip_runtime.h>
#include <hip/hip_bf16.h>
#include <math.h>

#define BATCH 4
#define SEQ   1024
#define DMODEL 1024
#define HEADS 16
#define DK    64

typedef float v2f __attribute__((ext_vector_type(2)));
typedef float v8f __attribute__((ext_vector_type(8)));
typedef unsigned int u32x4 __attribute__((ext_vector_type(4)));
typedef int i32x4 __attribute__((ext_vector_type(4)));
typedef int i32x8 __attribute__((ext_vector_type(8)));

__device__ __forceinline__ v8f wmma4(v2f a, v2f b, v8f c) {
  // D = A(16x4) x B(4x16) + C(16x16), fp32, wave32
  return __builtin_amdgcn_wmma_f32_16x16x4_f32(false, a, false, b, (short)0, c, false, false);
}

// ---------------------------------------------------------------------------
// Tensor Data Mover: DMA a 2D fp32 tile (tile_w x tile_h, row stride in
// elements) from global memory into LDS at byte offset lds_off, with optional
// hardware padding after every (1<<...) DWORDs to realize bank-friendly LDS
// strides.  D# layout per CDNA5 ISA ch.8.  pad_code: 4 -> pad after 32 DW,
// 5 -> pad after 64 DW; pad amount fixed at 1 DWORD.
// ---------------------------------------------------------------------------
__device__ __forceinline__ void tdm_load_2d(unsigned lds_off, const void* gptr,
                                            unsigned tile_w, unsigned tile_h,
                                            unsigned row_stride_elems,
                                            unsigned pad_code, unsigned pad_en) {
  const unsigned long long ga = (unsigned long long)(uintptr_t)gptr;
  u32x4 g0;
  g0.x = 1u;                                          // count=1, user desc
  g0.y = lds_off;                                     // lds_addr (bytes)
  g0.z = (unsigned)ga;                                // global_addr[31:0]
  g0.w = (unsigned)((ga >> 32) & 0x01FFFFFFu) | (2u << 30);  // addr[56:32], type=2
  i32x8 g1;
  g1[0] = (int)((2u << 16) | (pad_en << 20) | (pad_code << 22)); // 4B elems, padding
  g1[1] = (int)((tile_w & 0xFFFFu) << 16);            // tensor_dim0[15:0] @ bits63:48
  g1[2] = (int)(((tile_w >> 16) & 0xFFFFu) | ((tile_h & 0xFFFFu) << 16)); // d0 hi | d1 lo
  g1[3] = (int)(((tile_h >> 16) & 0xFFFFu) | ((tile_w & 0xFFFFu) << 16)); // d1 hi | tile_dim0
  g1[4] = (int)(tile_h & 0xFFFFu);                    // tile_dim1, tile_dim2=0 (2D)
  g1[5] = (int)row_stride_elems;                      // tensor_dim0_stride[31:0]
  g1[6] = 0;                                          // stride[47:32], dim1_stride lo
  g1[7] = 0;                                          // dim1_stride hi (unused, 2D)
  i32x4 gz = {0, 0, 0, 0};
#if __clang_major__ >= 23
  i32x8 gz8 = {0, 0, 0, 0, 0, 0, 0, 0};
  __builtin_amdgcn_tensor_load_to_lds(g0, g1, gz, gz, gz8, 0);
#else
  __builtin_amdgcn_tensor_load_to_lds(g0, g1, gz, gz, 0);
#endif
}

// ---------------------------------------------------------------------------
// Sinusoidal relative position embeddings: pe[t][c]=sin, pe[t][c+512]=cos
// positions run s-1 ... 0  (pos_seq[t] = S-1-t)
// ---------------------------------------------------------------------------
__global__ void posemb_kernel(float* __restrict__ pe) {
  const int t = blockIdx.x;                 // 0..SEQ-1
  const float ps = (float)(SEQ - 1 - t);
  const float kfac = logf(10000.0f) / 512.0f;
  for (int c = threadIdx.x; c < 512; c += blockDim.x) {
    const float invf = expf(-(float)c * kfac);
    const float x = ps * invf;
    pe[(size_t)t * DMODEL + c]       = sinf(x);
    pe[(size_t)t * DMODEL + 512 + c] = cosf(x);
  }
}

// ---------------------------------------------------------------------------
// C[M,N] = A[M,K] @ W[N,K]^T  (torch Linear).  M%128==0, N%64==0, K%32==0.
// Block: 256 threads = 8 waves. C-tile 128x64; wave w owns row strip w*16,
// 4 col sub-tiles. WMMA f32 16x16x4.  A/W chunks staged by the TDM with
// hardware padding to LDS stride 33.
// ---------------------------------------------------------------------------
__global__ __launch_bounds__(256) void gemm_nt(const float* __restrict__ A,
                                               const float* __restrict__ W,
                                               float* __restrict__ C,
                                               int M, int N, int K) {
  __shared__ float sA[128 * 33];
  __shared__ float sW[64 * 33];
  const int t = threadIdx.x;
  const int w = t >> 5, lane = t & 31, lr = lane & 15, half = lane >> 4;
  const int m0 = blockIdx.y * 128, n0 = blockIdx.x * 64;
  const unsigned ldsA = (unsigned)(uintptr_t)(void*)sA;
  const unsigned ldsW = (unsigned)(uintptr_t)(void*)sW;

  v8f acc[4] = {v8f{}, v8f{}, v8f{}, v8f{}};

  for (int k0 = 0; k0 < K; k0 += 32) {
    __syncthreads();
    if (w == 0) {
      tdm_load_2d(ldsA, A + (size_t)m0 * K + k0, 32u, 128u, (unsigned)K, 4u, 1u);
      tdm_load_2d(ldsW, W + (size_t)n0 * K + k0, 32u, 64u,  (unsigned)K, 4u, 1u);
      __builtin_amdgcn_s_wait_tensorcnt(0);
    }
    __syncthreads();
#pragma unroll
    for (int ks = 0; ks < 8; ++ks) {
      const int kk = ks * 4 + 2 * half;
      v2f a;
      a.x = sA[(w * 16 + lr) * 33 + kk];
      a.y = sA[(w * 16 + lr) * 33 + kk + 1];
#pragma unroll
      for (int cb = 0; cb < 4; ++cb) {
        v2f bf;                                   // B(k,n) = W[n][k]
        bf.x = sW[(cb * 16 + lr) * 33 + kk];
        bf.y = sW[(cb * 16 + lr) * 33 + kk + 1];
        acc[cb] = wmma4(a, bf, acc[cb]);
      }
    }
  }
#pragma unroll
  for (int cb = 0; cb < 4; ++cb)
#pragma unroll
    for (int r = 0; r < 8; ++r) {
      const int m = m0 + w * 16 + r + 8 * half;
      const int n = n0 + cb * 16 + lr;
      C[(size_t)m * N + n] = acc[cb][r];
    }
}

// ---------------------------------------------------------------------------
// Fused relative-position attention, flash style.
// grid = (SEQ/64, HEADS, BATCH), block = 256 (8 waves).
// Content score via WMMA; shifted pos score via windowed WMMA GEMM
//   G[r][x] = (q[r]+vb) . Pg[x],  Pg[x] = p[row(delta)], delta = x-63+(j0-i0)
//   row(delta) = S-1+delta (delta<=0) | ZERO (delta==1) | delta-2 (delta>=2)
//   shifted[a][col] = G[(delta<=0)?a:a+1][63+col-a]
// K/V tiles staged by the TDM (hardware-padded to LDS stride 65).
// ---------------------------------------------------------------------------
#define SQ_ST 65
#define SK_ST 65
#define SG_ST 129

__global__ __launch_bounds__(256) void attn_kernel(
    const float* __restrict__ Qg, const float* __restrict__ Kg,
    const float* __restrict__ Vg, const float* __restrict__ Pw,
    const float* __restrict__ ub, const float* __restrict__ vb,
    const unsigned char* __restrict__ maskg, float* __restrict__ ctx) {
  __shared__ float sQ[80 * SQ_ST];
  __shared__ float sU[64], sVb[64];
  __shared__ float sK[64 * SK_ST];
  __shared__ float sVt[64 * SK_ST];
  __shared__ float sPg[128 * SK_ST];
  __shared__ float sS[64 * SK_ST];
  __shared__ float sG[80 * SG_ST];
  __shared__ float sM[64], sL[64], sFac[64];
  __shared__ float mq[64], mk[64];

  const int t = threadIdx.x;
  const int w = t >> 5, lane = t & 31, lr = lane & 15, half = lane >> 4;
  const int i0 = blockIdx.x * 64;
  const int h = blockIdx.y, b = blockIdx.z;
  const size_t hoff = (size_t)h * DK;
  const float* qbase = Qg + ((size_t)b * SEQ) * DMODEL + hoff;
  const unsigned ldsK = (unsigned)(uintptr_t)(void*)sK;
  const unsigned ldsV = (unsigned)(uintptr_t)(void*)sVt;

  // Q rows i0..i0+64 (65 rows; rows >=65 zero pad for the 80-row G GEMM)
  for (int idx = t; idx < 80 * 64; idx += 256) {
    const int r = idx >> 6, c = idx & 63;
    float val = 0.0f;
    if (r < 65 && (i0 + r) < SEQ) val = qbase[(size_t)(i0 + r) * DMODEL + c];
    sQ[r * SQ_ST + c] = val;
  }
  if (t < 64) {
    sU[t] = ub[h * DK + t];
    sVb[t] = vb[h * DK + t];
    sM[t] = -3.0e38f;
    sL[t] = 0.0f;
    mq[t] = maskg[(size_t)b * SEQ + i0 + t] ? 1.0f : 0.0f;
  }

  const int ra = w >> 1, cb0 = (w & 1) * 2;   // this wave's O tiles
  v8f o0 = {}, o1 = {};

  for (int j0 = 0; j0 < SEQ; j0 += 64) {
    __syncthreads();
    // ---- stage K, V via TDM; Pg window + mask manually ----
    if (w == 0) {
      tdm_load_2d(ldsK, Kg + ((size_t)b * SEQ + j0) * DMODEL + hoff, 64u, 64u,
                  (unsigned)DMODEL, 5u, 1u);
      tdm_load_2d(ldsV, Vg + ((size_t)b * SEQ + j0) * DMODEL + hoff, 64u, 64u,
                  (unsigned)DMODEL, 5u, 1u);
    }
    const int dofs = j0 - i0;
    for (int idx = t; idx < 128 * 64; idx += 256) {
      const int x = idx >> 6, c = idx & 63;
      const int delta = x - 63 + dofs;
      float val = 0.0f;
      if (x < 127 && delta != 1) {
        const int prow = (delta <= 0) ? (SEQ - 1 + delta) : (delta - 2);
        val = Pw[(size_t)prow * DMODEL + hoff + c];
      }
      sPg[x * SK_ST + c] = val;
    }
    if (t < 64) mk[t] = maskg[(size_t)b * SEQ + j0 + t] ? 1.0f : 0.0f;
    if (w == 0) __builtin_amdgcn_s_wait_tensorcnt(0);
    __syncthreads();

    // ---- content score Sc = (Q+u) K^T : 16 sub-tiles, 2 per wave ----
#pragma unroll
    for (int s2 = 0; s2 < 2; ++s2) {
      const int id = w * 2 + s2;
      const int sra = id >> 2, scb = id & 3;
      v8f acc = {};
#pragma unroll
      for (int ks = 0; ks < 16; ++ks) {
        const int kk = ks * 4 + 2 * half;
        v2f a;
        a.x = sQ[(sra * 16 + lr) * SQ_ST + kk] + sU[kk];
        a.y = sQ[(sra * 16 + lr) * SQ_ST + kk + 1] + sU[kk + 1];
        v2f bf;                                   // B(k,n) = K[n][k]
        bf.x = sK[(scb * 16 + lr) * SK_ST + kk];
        bf.y = sK[(scb * 16 + lr) * SK_ST + kk + 1];
        acc = wmma4(a, bf, acc);
      }
#pragma unroll
      for (int r = 0; r < 8; ++r)
        sS[(sra * 16 + r + 8 * half) * SK_ST + scb * 16 + lr] = acc[r];
    }

    // ---- pos window G = (Q+vb) Pg^T : 80x128, 5 sub-tiles per wave ----
#pragma unroll
    for (int s5 = 0; s5 < 5; ++s5) {
      const int gr = s5, gx = w;
      v8f acc = {};
#pragma unroll
      for (int ks = 0; ks < 16; ++ks) {
        const int kk = ks * 4 + 2 * half;
        v2f a;
        a.x = sQ[(gr * 16 + lr) * SQ_ST + kk] + sVb[kk];
        a.y = sQ[(gr * 16 + lr) * SQ_ST + kk + 1] + sVb[kk + 1];
        v2f bf;                                   // B(k,x) = Pg[x][k]
        bf.x = sPg[(gx * 16 + lr) * SK_ST + kk];
        bf.y = sPg[(gx * 16 + lr) * SK_ST + kk + 1];
        acc = wmma4(a, bf, acc);
      }
#pragma unroll
      for (int r = 0; r < 8; ++r)
        sG[(gr * 16 + r + 8 * half) * SG_ST + gx * 16 + lr] = acc[r];
    }
    __syncthreads();

    // ---- combine + rel-shift + mask + online softmax ----
    {
      const int a = t >> 2, qd = t & 3;           // 4 lanes per row, same wave
      const float mqa = mq[a];
      float sc16[16];
      float tmax = -3.0e38f;
#pragma unroll
      for (int e = 0; e < 16; ++e) {
        const int col = qd * 16 + e;
        const int x = col - a + 63;               // 0..126
        const int delta = col - a + dofs;
        const int qa = (delta <= 0) ? a : (a + 1);
        const float sp = sG[qa * SG_ST + x];      // delta==1 row is zero
        float sc = (sS[a * SK_ST + col] + sp) * 0.125f;
        if (mqa == 0.0f || mk[col] == 0.0f) sc = -1.0e9f;
        sc16[e] = sc;
        tmax = fmaxf(tmax, sc);
      }
      tmax = fmaxf(tmax, __shfl_xor(tmax, 1, 32));
      tmax = fmaxf(tmax, __shfl_xor(tmax, 2, 32));
      const float mold = sM[a];
      const float mnew = fmaxf(mold, tmax);
      const float corr = expf(mold - mnew);
      float lsum = 0.0f;
#pragma unroll
      for (int e = 0; e < 16; ++e) {
        const float pv = expf(sc16[e] - mnew);
        sS[a * SK_ST + qd * 16 + e] = pv;         // overwrite Sc with P~
        lsum += pv;
      }
      lsum += __shfl_xor(lsum, 1, 32);
      lsum += __shfl_xor(lsum, 2, 32);
      if (qd == 0) {
        sM[a] = mnew;
        sL[a] = sL[a] * corr + lsum;
        sFac[a] = corr;
      }
    }
    __syncthreads();

    // ---- O = O*corr + P~ . V  (WMMA) ----
#pragma unroll
    for (int r = 0; r < 8; ++r) {
      const float f = sFac[ra * 16 + r + 8 * half];
      o0[r] *= f;
      o1[r] *= f;
    }
#pragma unroll
    for (int ks = 0; ks < 16; ++ks) {
      const int kk = ks * 4 + 2 * half;
      v2f a;
      a.x = sS[(ra * 16 + lr) * SK_ST + kk];
      a.y = sS[(ra * 16 + lr) * SK_ST + kk + 1];
      v2f b0;                                     // B(k,n) = V[k][n]
      b0.x = sVt[kk * SK_ST + cb0 * 16 + lr];
      b0.y = sVt[(kk + 1) * SK_ST + cb0 * 16 + lr];
      o0 = wmma4(a, b0, o0);
      v2f b1;
      b1.x = sVt[kk * SK_ST + (cb0 + 1) * 16 + lr];
      b1.y = sVt[(kk + 1) * SK_ST + (cb0 + 1) * 16 + lr];
      o1 = wmma4(a, b1, o1);
    }
  }

  // ---- normalize and write context (b, q, h, dk) ----
  float* cbase = ctx + ((size_t)b * SEQ) * DMODEL + hoff;
#pragma unroll
  for (int r = 0; r < 8; ++r) {
    const int m = ra * 16 + r + 8 * half;
    const float linv = 1.0f / sL[m];
    const int gi = i0 + m;
    cbase[(size_t)gi * DMODEL + cb0 * 16 + lr] = o0[r] * linv;
    cbase[(size_t)gi * DMODEL + (cb0 + 1) * 16 + lr] = o1[r] * linv;
  }
}

// ---------------------------------------------------------------------------
extern "C" void kernel_launch(void* const* d_in, const int* in_sizes, int n_in,
                              void* d_out, int out_size, void* d_ws, size_t ws_size,
                              hipStream_t stream) {
  (void)in_sizes; (void)n_in; (void)out_size; (void)ws_size;
  const float* query = (const float*)d_in[0];
  const float* key   = (const float*)d_in[1];
  const float* value = (const float*)d_in[2];
  const unsigned char* mask = (const unsigned char*)d_in[3];
  const float* Wq = (const float*)d_in[4];
  const float* Wk = (const float*)d_in[5];
  const float* Wv = (const float*)d_in[6];
  const float* Wp = (const float*)d_in[7];
  const float* Wo = (const float*)d_in[8];
  const float* pbu = (const float*)d_in[9];
  const float* pbv = (const float*)d_in[10];
  float* out = (float*)d_out;

  // workspace layout (floats): pe 1M | q 4M | k 4M | v 4M | p 1M | ctx 4M = 72MB
  float* ws = (float*)d_ws;
  float* pe  = ws;
  float* qw  = ws + (1u << 20);
  float* kw  = ws + (5u << 20);
  float* vw  = ws + (9u << 20);
  float* pw  = ws + (13u << 20);
  float* cw  = ws + (14u << 20);

  const int MQ = BATCH * SEQ;  // 4096

  posemb_kernel<<<SEQ, 256, 0, stream>>>(pe);

  gemm_nt<<<dim3(DMODEL / 64, MQ / 128), 256, 0, stream>>>(query, Wq, qw, MQ, DMODEL, DMODEL);
  gemm_nt<<<dim3(DMODEL / 64, MQ / 128), 256, 0, stream>>>(key,   Wk, kw, MQ, DMODEL, DMODEL);
  gemm_nt<<<dim3(DMODEL / 64, MQ / 128), 256, 0, stream>>>(value, Wv, vw, MQ, DMODEL, DMODEL);
  gemm_nt<<<dim3(DMODEL / 64, SEQ / 128), 256, 0, stream>>>(pe,   Wp, pw, SEQ, DMODEL, DMODEL);

  attn_kernel<<<dim3(SEQ / 64, HEADS, BATCH), 256, 0, stream>>>(
      qw, kw, vw, pw, pbu, pbv, mask, cw);

  gemm_nt<<<dim3(DMODEL / 64, MQ / 128), 256, 0, stream>>>(cw, Wo, out, MQ, DMODEL, DMODEL);
}